// UnfoldPlainFit_42477226558039
// MI455X (gfx1250) — compile-verified
//
#include <hip/hip_runtime.h>
#include <stdint.h>

// Plane-fit over 7x7 neighborhoods == three fixed separable convolutions.
//   C0 = (1/196) * [vertical box-sum] x [horizontal ramp -3..3]
//   C1 = (1/196) * [vertical ramp]    x [horizontal box-sum]
//   C2 = (1/49)  * 7x7 box mean
// Bandwidth-bound: 64 MB min traffic -> ~2.8 us at 23.3 TB/s. gfx1250 path:
// async global->LDS halo staging (ASYNCcnt, b128 wide for interior tiles)
// + NT stores for the streaming output.

#define KS     7
#define PADR   3
#define IMG_H  2048
#define IMG_W  2048

#define TW     64              // output tile width
#define TH     32              // output tile height
#define LH     (TH + KS - 1)   // 38 halo rows
#define LSTR   72              // LDS row = 72 columns, col c <-> gx = bx-4+c
#define NVEC   (LSTR / 4)      // 18 float4 segments per row
#define NTHR   256

typedef int v4i __attribute__((ext_vector_type(4)));
typedef __attribute__((address_space(1))) int gint_t;    // global int
typedef __attribute__((address_space(3))) int lint_t;    // LDS int
typedef __attribute__((address_space(1))) v4i gv4i_t;    // global int4
typedef __attribute__((address_space(3))) v4i lv4i_t;    // LDS int4

__device__ __forceinline__ void async_g2l_b32(const float* gsrc, float* lds_dst) {
#if __has_builtin(__builtin_amdgcn_global_load_async_to_lds_b32)
    __builtin_amdgcn_global_load_async_to_lds_b32(
        (gint_t*)(void*)gsrc, (lint_t*)(void*)lds_dst, 0, 0);
#else
    *lds_dst = *gsrc;
#endif
}

__device__ __forceinline__ void async_g2l_b128(const float* gsrc, float* lds_dst) {
#if __has_builtin(__builtin_amdgcn_global_load_async_to_lds_b128)
    __builtin_amdgcn_global_load_async_to_lds_b128(
        (gv4i_t*)(void*)gsrc, (lv4i_t*)(void*)lds_dst, 0, 0);
#else
    lds_dst[0] = gsrc[0]; lds_dst[1] = gsrc[1];
    lds_dst[2] = gsrc[2]; lds_dst[3] = gsrc[3];
#endif
}

__device__ __forceinline__ void wait_async_zero() {
#if __has_builtin(__builtin_amdgcn_s_wait_asynccnt)
    __builtin_amdgcn_s_wait_asynccnt(0);
#else
    asm volatile("s_wait_asynccnt 0x0" ::: "memory");
#endif
}

__global__ __launch_bounds__(NTHR)
void plane_fit7x7_kernel(const float* __restrict__ x, float* __restrict__ out) {
    __shared__ float s_in[LH * LSTR];  // halo input tile   (~10.9 KB)
    __shared__ float s_hs[LH * TW];    // horizontal box-sum (~9.5 KB)
    __shared__ float s_hr[LH * TW];    // horizontal ramp    (~9.5 KB)

    const int tid = threadIdx.x;
    const int bx  = blockIdx.x * TW;
    const int by  = blockIdx.y * TH;

    // ---- Stage A: async-stage the 38x72 halo window into LDS ---------------
    // Layout: s_in[ly*LSTR + c] = x[clamp(by-3+ly), clamp(bx-4+c)]
    const bool interior = (bx >= 4) && (bx + (LSTR - 4) <= IMG_W) &&
                          (by >= PADR) && (by + LH - PADR <= IMG_H);
    if (interior) {
        // Whole window in-bounds; 16B-aligned both sides -> wide b128 async.
        const float* base = x + (size_t)(by - PADR) * IMG_W + (bx - 4);
        for (int t = tid; t < LH * NVEC; t += NTHR) {
            int ly  = t / NVEC;            // 0..37
            int seg = t - ly * NVEC;       // 0..17
            async_g2l_b128(base + (size_t)ly * IMG_W + seg * 4,
                           &s_in[ly * LSTR + seg * 4]);
        }
    } else {
        // Border tiles: per-element clamped (edge-replicate) b32 async gather.
        for (int t = tid; t < LH * LSTR; t += NTHR) {
            int ly = t / LSTR;
            int c  = t - ly * LSTR;
            int gx = bx - 4 + c;
            int gy = by - PADR + ly;
            gx = gx < 0 ? 0 : (gx > IMG_W - 1 ? IMG_W - 1 : gx);
            gy = gy < 0 ? 0 : (gy > IMG_H - 1 ? IMG_H - 1 : gy);
            async_g2l_b32(&x[(size_t)gy * IMG_W + gx], &s_in[t]);
        }
    }
    wait_async_zero();
    __syncthreads();

    // ---- Stage B: horizontal 7-tap box-sum + ramp-sum per row --------------
    // column lx of the output tile reads s_in columns lx+1 .. lx+7
    // ramp = -3e0 -2e1 -e2 +e4 +2e5 +3e6 = 3(e6-e0) + 2(e5-e1) + (e4-e2)
    for (int t = tid; t < LH * TW; t += NTHR) {
        int ly = t >> 6;            // TW == 64
        int lx = t & (TW - 1);
        const float* r = &s_in[ly * LSTR + lx + 1];
        float e0 = r[0], e1 = r[1], e2 = r[2], e3 = r[3];
        float e4 = r[4], e5 = r[5], e6 = r[6];
        float s = ((e0 + e6) + (e1 + e5)) + ((e2 + e4) + e3);
        float g = fmaf(3.0f, e6 - e0, fmaf(2.0f, e5 - e1, e4 - e2));
        s_hs[t] = s;
        s_hr[t] = g;
    }
    __syncthreads();

    // ---- Stage C: vertical 7-tap combine -> 3 plane coefficients -----------
    for (int t = tid; t < TH * TW; t += NTHR) {
        int ly = t >> 6;
        int lx = t & (TW - 1);
        const float* cs = &s_hs[ly * TW + lx];
        const float* cr = &s_hr[ly * TW + lx];

        float h0 = cs[0 * TW], h1 = cs[1 * TW], h2 = cs[2 * TW], h3 = cs[3 * TW];
        float h4 = cs[4 * TW], h5 = cs[5 * TW], h6 = cs[6 * TW];
        float r0 = cr[0 * TW], r1 = cr[1 * TW], r2 = cr[2 * TW], r3 = cr[3 * TW];
        float r4 = cr[4 * TW], r5 = cr[5 * TW], r6 = cr[6 * TW];

        // C0: vertical box-sum of horizontal ramp
        float c0 = (((r0 + r6) + (r1 + r5)) + ((r2 + r4) + r3)) * (1.0f / 196.0f);
        // C1: vertical ramp of horizontal box-sum
        float c1 = fmaf(3.0f, h6 - h0, fmaf(2.0f, h5 - h1, h4 - h2)) * (1.0f / 196.0f);
        // C2: 7x7 box mean
        float c2 = (((h0 + h6) + (h1 + h5)) + ((h2 + h4) + h3)) * (1.0f / 49.0f);

        size_t o = ((size_t)(by + ly) * IMG_W + (size_t)(bx + lx)) * 3;
        __builtin_nontemporal_store(c0, &out[o + 0]);   // streaming 48 MB output:
        __builtin_nontemporal_store(c1, &out[o + 1]);   // NT hint, keep L2 for the
        __builtin_nontemporal_store(c2, &out[o + 2]);   // reused input halo lines
    }
}

extern "C" void kernel_launch(void* const* d_in, const int* in_sizes, int n_in,
                              void* d_out, int out_size, void* d_ws, size_t ws_size,
                              hipStream_t stream) {
    const float* x = (const float*)d_in[0];
    float* out = (float*)d_out;
    dim3 grid(IMG_W / TW, IMG_H / TH);   // 32 x 64 = 2048 blocks
    plane_fit7x7_kernel<<<grid, NTHR, 0, stream>>>(x, out);
}